// TokenSparse_5523327942953
// MI455X (gfx1250) — compile-verified
//
#include <hip/hip_runtime.h>
#include <hip/hip_bf16.h>

#define AS1 __attribute__((address_space(1)))
#define AS3 __attribute__((address_space(3)))

typedef int v4i __attribute__((vector_size(16)));   // matches builtin's V4i pointee

#if defined(__has_builtin)
#  if __has_builtin(__builtin_amdgcn_global_load_async_to_lds_b128)
#    define HAVE_ASYNC_COPY 1
#  endif
#  if __has_builtin(__builtin_amdgcn_s_wait_asynccnt)
#    define HAVE_WAIT_ASYNC 1
#  endif
#endif

// Problem geometry (fixed by the reference): B=32, N=4096, C=768
constexpr int N_TOK   = 4096;
constexpr int THREADS = 256;           // 8 wave32 waves per block
constexpr int ELEMS   = N_TOK / THREADS; // 16 keys per thread

// ---------------------------------------------------------------------------
// Kernel 1: per-batch-row score = avg of 3 min-max-normalized attentions,
// radix-select the k-th largest, emit 0/1 mask (ties kept by ascending index,
// matching jax.lax.top_k stability).
// ---------------------------------------------------------------------------
__global__ __launch_bounds__(THREADS)
void score_topk_mask_kernel(const float* __restrict__ s1,
                            const float* __restrict__ s2,
                            const float* __restrict__ s3,
                            float* __restrict__ mask_out,
                            int num_keep)
{
    const int b = blockIdx.x;
    const int t = threadIdx.x;

    const float* arrs[3] = { s1 + (size_t)b * N_TOK,
                             s2 + (size_t)b * N_TOK,
                             s3 + (size_t)b * N_TOK };

    __shared__ float    sred_min[THREADS];
    __shared__ float    sred_max[THREADS];
    __shared__ float    gmin[3], gmax[3];
    __shared__ unsigned keys[N_TOK];       // 16 KB
    __shared__ int      hist[256];
    __shared__ unsigned sel_prefix;
    __shared__ int      sel_rem;

    // ---- per-array min / max over the row -------------------------------
    #pragma unroll
    for (int a = 0; a < 3; ++a) {
        float mn =  3.4e38f, mx = -3.4e38f;
        #pragma unroll
        for (int i = 0; i < ELEMS; ++i) {
            float v = arrs[a][t + i * THREADS];
            mn = fminf(mn, v);
            mx = fmaxf(mx, v);
        }
        sred_min[t] = mn; sred_max[t] = mx;
        __syncthreads();
        for (int s = THREADS / 2; s > 0; s >>= 1) {
            if (t < s) {
                sred_min[t] = fminf(sred_min[t], sred_min[t + s]);
                sred_max[t] = fmaxf(sred_max[t], sred_max[t + s]);
            }
            __syncthreads();
        }
        if (t == 0) { gmin[a] = sred_min[0]; gmax[a] = sred_max[0]; }
        __syncthreads();
    }

    float mn0 = gmin[0], mn1 = gmin[1], mn2 = gmin[2];
    float i0 = 1.0f / (gmax[0] - mn0 + 1e-8f);
    float i1 = 1.0f / (gmax[1] - mn1 + 1e-8f);
    float i2 = 1.0f / (gmax[2] - mn2 + 1e-8f);

    // ---- score -> order-preserving u32 key ------------------------------
    #pragma unroll
    for (int i = 0; i < ELEMS; ++i) {
        int n = t + i * THREADS;
        float sc = ((arrs[0][n] - mn0) * i0 +
                    (arrs[1][n] - mn1) * i1 +
                    (arrs[2][n] - mn2) * i2) * (1.0f / 3.0f);
        unsigned u = __float_as_uint(sc);
        keys[n] = (u & 0x80000000u) ? ~u : (u | 0x80000000u);
    }
    __syncthreads();

    // ---- 4-pass radix select of the num_keep-th largest key -------------
    unsigned prefix = 0;
    int remaining = num_keep;
    for (int pass = 0; pass < 4; ++pass) {
        const int shift = 24 - pass * 8;
        hist[t] = 0;                       // THREADS == 256 bins
        __syncthreads();
        for (int i = 0; i < ELEMS; ++i) {
            unsigned k = keys[t + i * THREADS];
            bool match = (pass == 0) || ((k >> (shift + 8)) == prefix);
            if (match) atomicAdd(&hist[(k >> shift) & 0xFFu], 1);
        }
        __syncthreads();
        if (t == 0) {
            int rem = remaining; int d = 0;
            for (int bin = 255; bin >= 0; --bin) {
                int c = hist[bin];
                if (rem <= c) { d = bin; break; }
                rem -= c;
            }
            sel_prefix = (prefix << 8) | (unsigned)d;
            sel_rem    = rem;
        }
        __syncthreads();
        prefix = sel_prefix; remaining = sel_rem;
        __syncthreads();
    }
    const unsigned T = prefix;             // key of the k-th largest element

    // ---- emit mask ------------------------------------------------------
    float* mout = mask_out + (size_t)b * N_TOK;
    #pragma unroll
    for (int i = 0; i < ELEMS; ++i) {
        int n = t + i * THREADS;
        unsigned k = keys[n];
        if (k != T) mout[n] = (k > T) ? 1.0f : 0.0f;   // equals handled below
    }
    __syncthreads();
    if (t == 0) {                          // ties: keep smallest indices first
        int q = remaining;
        for (int n = 0; n < N_TOK; ++n) {
            if (keys[n] == T) { mout[n] = (q > 0) ? 1.0f : 0.0f; if (q > 0) --q; }
        }
    }
}

// ---------------------------------------------------------------------------
// Kernel 2: masked_tokens = tokens * mask[row].  Pure streaming @ HBM roofline.
// 4 rows (12 KB) per block, staged via CDNA5 async global->LDS copies.
// ---------------------------------------------------------------------------
constexpr int F4_PER_ROW    = 192;  // 768 floats / 4
constexpr int ROWS_PER_BLK  = 4;
constexpr int TILE_F4       = F4_PER_ROW * ROWS_PER_BLK; // 768 float4 = 12 KB

__global__ __launch_bounds__(256)
void mask_mul_kernel(const float4* __restrict__ tok,
                     const float*  __restrict__ mask,
                     float4* __restrict__ out)
{
    __shared__ float4 tile[TILE_F4];
    const int    t        = threadIdx.x;
    const size_t base     = (size_t)blockIdx.x * TILE_F4;
    const int    row_base = blockIdx.x * ROWS_PER_BLK;

#ifdef HAVE_ASYNC_COPY
    #pragma unroll
    for (int j = 0; j < 3; ++j) {
        int idx = t + j * 256;
        __builtin_amdgcn_global_load_async_to_lds_b128(
            (AS1 v4i*)(tok + base + idx),
            (AS3 v4i*)&tile[idx],
            /*offset=*/0, /*cpol=*/0);
    }
#ifdef HAVE_WAIT_ASYNC
    __builtin_amdgcn_s_wait_asynccnt(0);
#else
    asm volatile("s_wait_asynccnt 0" ::: "memory");
#endif
    __syncthreads();
#else
    #pragma unroll
    for (int j = 0; j < 3; ++j) {
        int idx = t + j * 256;
        tile[idx] = tok[base + idx];
    }
    __syncthreads();
#endif

    #pragma unroll
    for (int j = 0; j < 3; ++j) {
        int idx = t + j * 256;
        float s = mask[row_base + idx / F4_PER_ROW];
        float4 v = tile[idx];
        v.x *= s; v.y *= s; v.z *= s; v.w *= s;
        out[base + idx] = v;
    }
}

// ---------------------------------------------------------------------------
extern "C" void kernel_launch(void* const* d_in, const int* in_sizes, int n_in,
                              void* d_out, int out_size, void* d_ws, size_t ws_size,
                              hipStream_t stream)
{
    (void)n_in; (void)out_size; (void)d_ws; (void)ws_size;

    const float* tokens = (const float*)d_in[0];
    const float* s1     = (const float*)d_in[1];
    const float* s2     = (const float*)d_in[2];
    const float* s3     = (const float*)d_in[3];

    const int BN = in_sizes[1];            // B * N  (N fixed at 4096)
    const int B  = BN / N_TOK;
    const int C  = in_sizes[0] / BN;       // 768
    (void)C;

    float* out_tokens = (float*)d_out;
    float* out_mask   = (float*)d_out + (size_t)in_sizes[0];

    int num_keep = (6 * N_TOK + 9) / 10;   // ceil(0.6 * N) = 2458
    if (num_keep < 1) num_keep = 1;

    score_topk_mask_kernel<<<B, THREADS, 0, stream>>>(s1, s2, s3, out_mask, num_keep);

    const int grid2 = BN / ROWS_PER_BLK;   // 32768 blocks of 4 rows
    mask_mul_kernel<<<grid2, 256, 0, stream>>>(
        (const float4*)tokens, out_mask, (float4*)out_tokens);
}